// BasicConvolutionBlock_257698038139
// MI455X (gfx1250) — compile-verified
//
#include <hip/hip_runtime.h>
#include <stdint.h>

typedef float v2f __attribute__((ext_vector_type(2)));
typedef float v8f __attribute__((ext_vector_type(8)));

#define N_VOX   100000
#define C_IN    64
#define C_OUT   64
#define K3      27
#define MPAIR   40000
#define TILES_PER_K (MPAIR / 16)      // 2500 tiles of 16 pairs
#define WAVES_PER_BLOCK 8
#define EPS_BN  1e-5f

// Hardware FP32 atomic add at device scope (avoid CAS-loop lowering).
__device__ __forceinline__ void atomic_add_f32(float* p, float v) {
    asm volatile("global_atomic_add_f32 %0, %1, off scope:SCOPE_DEV"
                 :: "v"((uint64_t)(uintptr_t)p), "v"(v) : "memory");
}

// ---------------------------------------------------------------------------
// Kernel 0: zero the accumulator (d_out) and the stats scratch (sum/sumsq).
// ---------------------------------------------------------------------------
__global__ void zero_kernel(float* __restrict__ out, float* __restrict__ stats, int total) {
    int i = blockIdx.x * blockDim.x + threadIdx.x;
    if (i < total) out[i] = 0.0f;
    if (i < 128)   stats[i] = 0.0f;
}

// ---------------------------------------------------------------------------
// Kernel 1: gather -> WMMA fp32 GEMM -> scatter-add.
// grid = (ceil(2500/8), 27); block = 256 threads = 8 wave32s; 1 tile per wave.
// ---------------------------------------------------------------------------
__global__ void __launch_bounds__(256)
conv_wmma_kernel(const float* __restrict__ x, const float* __restrict__ W,
                 const int* __restrict__ in_idx, const int* __restrict__ out_idx,
                 float* __restrict__ out) {
    __shared__ float sW[C_IN * C_OUT];   // transposed: sW[co*64 + ci]

    const int k   = blockIdx.y;          // kernel offset 0..26
    const int tid = threadIdx.x;

    // Cooperative transposed load of W[k] (64x64 fp32 = 16 KB of 320 KB LDS).
    const float* Wk = W + k * (C_IN * C_OUT);
    for (int i = tid; i < C_IN * C_OUT; i += 256) {
        int ci = i >> 6, co = i & 63;
        sW[co * 64 + ci] = Wk[i];
    }
    __syncthreads();

    const int wave = tid >> 5;
    const int lane = tid & 31;
    const int tile = blockIdx.x * WAVES_PER_BLOCK + wave;
    if (tile >= TILES_PER_K) return;     // uniform per-wave: EXEC stays all-1s

    const int row_base = tile * 16;
    const int m    = lane & 15;          // A-row / N-col owned by this lane
    const int koff = (lane >> 4) << 1;   // K sub-offset 0 or 2 (ISA A/B layout)

    // ---- Gather A tile: 16 rows x 64 channels, as 16 float2 per lane ----
    const int g_in = in_idx[k * MPAIR + row_base + m];
    const float* xr = x + (size_t)g_in * C_IN + koff;
    v2f a[16];
#pragma unroll
    for (int kk = 0; kk < 16; ++kk)
        a[kk] = *(const v2f*)(xr + kk * 4);   // 8B-aligned global_load_b64

    // ---- 64 x V_WMMA_F32_16X16X4_F32: K=64 in 16 steps, 4 N-tiles ----
    v8f c0 = {}, c1 = {}, c2 = {}, c3 = {};
#pragma unroll
    for (int kk = 0; kk < 16; ++kk) {
        const int kb = kk * 4 + koff;
        v2f b0 = *(const v2f*)(sW + ( 0 + m) * 64 + kb);  // ds_load_b64
        v2f b1 = *(const v2f*)(sW + (16 + m) * 64 + kb);
        v2f b2 = *(const v2f*)(sW + (32 + m) * 64 + kb);
        v2f b3 = *(const v2f*)(sW + (48 + m) * 64 + kb);
        c0 = __builtin_amdgcn_wmma_f32_16x16x4_f32(false, a[kk], false, b0, (short)0, c0, false, false);
        c1 = __builtin_amdgcn_wmma_f32_16x16x4_f32(false, a[kk], false, b1, (short)0, c1, false, false);
        c2 = __builtin_amdgcn_wmma_f32_16x16x4_f32(false, a[kk], false, b2, (short)0, c2, false, false);
        c3 = __builtin_amdgcn_wmma_f32_16x16x4_f32(false, a[kk], false, b3, (short)0, c3, false, false);
    }

    // ---- Scatter-add: C/D layout -> VGPR r holds M = r + 8*(lane>>4), N = lane&15
    const int half = (lane >> 4) << 3;
    int oidx[8];
#pragma unroll
    for (int r = 0; r < 8; ++r)
        oidx[r] = out_idx[k * MPAIR + row_base + half + r];
#pragma unroll
    for (int r = 0; r < 8; ++r) {
        float* orow = out + (size_t)oidx[r] * C_OUT + m;
        atomic_add_f32(orow +  0, c0[r]);
        atomic_add_f32(orow + 16, c1[r]);
        atomic_add_f32(orow + 32, c2[r]);
        atomic_add_f32(orow + 48, c3[r]);
    }
}

// ---------------------------------------------------------------------------
// Kernel 2: per-channel sum / sum-of-squares reduction over N_VOX rows.
// stride is a multiple of 64, so each thread owns a fixed channel (tid & 63).
// ---------------------------------------------------------------------------
__global__ void __launch_bounds__(256)
stats_kernel(const float* __restrict__ out, float* __restrict__ stats, int total) {
    __shared__ float ssum[C_OUT];
    __shared__ float ssq[C_OUT];
    const int tid = threadIdx.x;
    if (tid < C_OUT) { ssum[tid] = 0.0f; ssq[tid] = 0.0f; }
    __syncthreads();

    float s = 0.0f, q = 0.0f;
    const int stride = gridDim.x * blockDim.x;     // multiple of 64
    for (int i = blockIdx.x * blockDim.x + tid; i < total; i += stride) {
        float v = out[i];
        s += v;
        q += v * v;
    }
    const int ch = tid & 63;
    atomicAdd(&ssum[ch], s);                       // ds_add_f32
    atomicAdd(&ssq[ch],  q);
    __syncthreads();
    if (tid < C_OUT) {
        atomic_add_f32(stats + tid,         ssum[tid]);
        atomic_add_f32(stats + C_OUT + tid, ssq[tid]);
    }
}

// ---------------------------------------------------------------------------
// Kernel 3: BatchNorm (biased var) + affine + ReLU, in place on d_out.
// ---------------------------------------------------------------------------
__global__ void __launch_bounds__(256)
bn_relu_kernel(float* __restrict__ out, const float* __restrict__ stats,
               const float* __restrict__ gamma, const float* __restrict__ beta,
               int total) {
    int i = blockIdx.x * blockDim.x + threadIdx.x;
    if (i >= total) return;
    const int ch = i & 63;
    const float inv_n = 1.0f / (float)N_VOX;
    float mean = stats[ch] * inv_n;
    float var  = stats[C_OUT + ch] * inv_n - mean * mean;
    float inv  = rsqrtf(var + EPS_BN);
    float v    = (out[i] - mean) * inv * gamma[ch] + beta[ch];
    out[i] = fmaxf(v, 0.0f);
}

// ---------------------------------------------------------------------------
extern "C" void kernel_launch(void* const* d_in, const int* in_sizes, int n_in,
                              void* d_out, int out_size, void* d_ws, size_t ws_size,
                              hipStream_t stream) {
    const float* x      = (const float*)d_in[0];
    const float* W      = (const float*)d_in[1];
    const float* gamma  = (const float*)d_in[2];
    const float* beta   = (const float*)d_in[3];
    const int*   in_idx = (const int*)d_in[4];
    const int*   out_idx= (const int*)d_in[5];
    float* out   = (float*)d_out;
    float* stats = (float*)d_ws;               // 128 floats: sum[64], sumsq[64]

    const int total = N_VOX * C_OUT;           // == out_size

    zero_kernel<<<(total + 255) / 256, 256, 0, stream>>>(out, stats, total);

    dim3 grid((TILES_PER_K + WAVES_PER_BLOCK - 1) / WAVES_PER_BLOCK, K3);
    conv_wmma_kernel<<<grid, 256, 0, stream>>>(x, W, in_idx, out_idx, out);

    stats_kernel<<<1024, 256, 0, stream>>>(out, stats, total);

    bn_relu_kernel<<<(total + 255) / 256, 256, 0, stream>>>(out, stats, gamma, beta, total);
}